// FrangiFilter_46377056862499
// MI455X (gfx1250) — compile-verified
//
#include <hip/hip_runtime.h>
#include <math.h>

typedef float v2f __attribute__((ext_vector_type(2)));
typedef float v8f __attribute__((ext_vector_type(8)));

#define W2   2048
#define NPIX (2048*2048)
#define KP   324      // padded composed-kernel stride: 16 + 289 + 16 -> 324
#define SSTRIDE 420   // LDS strip row stride >= 128 + 4*72 = 416

__device__ __forceinline__ int reflect_idx(int i, int n) {
    if (i < 0)  i = -i - 1;
    if (i >= n) i = 2*n - 1 - i;
    return i;
}

// ---------------- grayscale + negate, written transposed ----------------
__global__ __launch_bounds__(256)
void k_gray_t(const float* __restrict__ img, float* __restrict__ grayT) {
    __shared__ float tile[32][33];
    const int bx = blockIdx.x * 32, by = blockIdx.y * 32;
    const int tx = threadIdx.x, ty = threadIdx.y;       // 32 x 8
    for (int i = 0; i < 32; i += 8) {
        int y = by + ty + i, x = bx + tx;
        float r = img[(size_t)0*NPIX + (size_t)y*W2 + x];
        float g = img[(size_t)1*NPIX + (size_t)y*W2 + x];
        float b = img[(size_t)2*NPIX + (size_t)y*W2 + x];
        tile[ty + i][tx] = -(0.2989f*r + 0.587f*g + 0.114f*b);
    }
    __syncthreads();
    for (int i = 0; i < 32; i += 8) {
        int ro = ty + i, co = tx;
        grayT[(size_t)(bx + ro)*W2 + by + co] = tile[co][ro];
    }
}

// ---------------- composed Gaussian-derivative kernels (f64) ----------------
__global__ void k_make_kernels(float* __restrict__ kbuf, int sigma, int r) {
    __shared__ double sphi[160], sdrv[160];
    __shared__ double ssum;
    const int K1 = 2*r + 1, Klen = 4*r + 1;
    const int t = threadIdx.x;
    const double s2 = (double)sigma * (double)sigma;
    for (int i = t; i < K1; i += blockDim.x) {
        double x = (double)(i - r);
        sphi[i] = exp(-0.5 * x * x / s2);
    }
    __syncthreads();
    if (t == 0) { double s = 0.0; for (int i = 0; i < K1; ++i) s += sphi[i]; ssum = s; }
    __syncthreads();
    for (int i = t; i < K1; i += blockDim.x) {
        double p = sphi[i] / ssum;
        sphi[i] = p;
        sdrv[i] = ((double)(i - r) / s2) * p;
    }
    __syncthreads();
    for (int i = t; i < 3*KP; i += blockDim.x) kbuf[i] = 0.f;
    __syncthreads();
    for (int i = t; i < Klen; i += blockDim.x) {
        int jlo = max(0, i - (K1 - 1)), jhi = min(K1 - 1, i);
        double dd = 0.0, ds = 0.0, ss = 0.0;
        for (int j = jlo; j <= jhi; ++j) {
            dd += sdrv[j] * sdrv[i - j];
            ds += sdrv[j] * sphi[i - j];
            ss += sphi[j] * sphi[i - j];
        }
        kbuf[0*KP + 16 + i] = (float)dd;
        kbuf[1*KP + 16 + i] = (float)ds;
        kbuf[2*KP + 16 + i] = (float)ss;
    }
}

// ---------------- pass 1: vertical conv (horizontal on grayT), WMMA ----------------
__global__ __launch_bounds__(256)
void k_conv_v(const float* __restrict__ grayT, const float* __restrict__ kbuf,
              float* __restrict__ odd, float* __restrict__ ods, float* __restrict__ oss,
              int r) {
    __shared__ float sIn[16 * SSTRIDE];
    __shared__ float skp[3 * KP];
    __shared__ float sT[8 * 16 * 20];

    const int R = 2*r;
    const int SW = 128 + 4*r;
    const int chunks = r + 4;
    const int tid = threadIdx.x;
    const int wv = tid >> 5, lane = tid & 31, half = lane >> 4, ln = lane & 15;
    const int rowBase  = blockIdx.y * 16;     // grayT rows
    const int colBase0 = blockIdx.x * 128;    // grayT cols (block)

    for (int i = tid; i < 3*KP; i += 256) skp[i] = kbuf[i];
    for (int rr = 0; rr < 16; ++rr) {
        const float* src = grayT + (size_t)(rowBase + rr) * W2;
        for (int c = tid; c < SW; c += 256)
            sIn[rr*SSTRIDE + c] = src[reflect_idx(colBase0 - R + c, W2)];
    }
    __syncthreads();

    v8f c0 = {}, c1 = {}, c2 = {};
    const float* arow = &sIn[ln * SSTRIDE + wv*16 + 2*half];
    const int kb0 = 16 + 2*half - ln;
    for (int kc = 0; kc < chunks; ++kc) {
        v2f a;  a.x  = arow[kc*4];              a.y  = arow[kc*4 + 1];
        int kb = kb0 + kc*4;
        v2f b0; b0.x = skp[0*KP + kb];          b0.y = skp[0*KP + kb + 1];
        v2f b1; b1.x = skp[1*KP + kb];          b1.y = skp[1*KP + kb + 1];
        v2f b2; b2.x = skp[2*KP + kb];          b2.y = skp[2*KP + kb + 1];
        c0 = __builtin_amdgcn_wmma_f32_16x16x4_f32(false, a, false, b0, (short)0, c0, false, false);
        c1 = __builtin_amdgcn_wmma_f32_16x16x4_f32(false, a, false, b1, (short)0, c1, false, false);
        c2 = __builtin_amdgcn_wmma_f32_16x16x4_f32(false, a, false, b2, (short)0, c2, false, false);
    }

    float* outs[3] = {odd, ods, oss};
    v8f accs[3] = {c0, c1, c2};
    const int cB = colBase0 + wv*16;   // grayT col of this tile == gray row
    #pragma unroll
    for (int k = 0; k < 3; ++k) {
        __syncthreads();
        float* tb = &sT[(wv*16 + ln) * 20];
        #pragma unroll
        for (int vg = 0; vg < 8; ++vg) tb[vg + 8*half] = accs[k][vg];
        __syncthreads();
        #pragma unroll
        for (int i = 0; i < 2; ++i) {
            int m0 = half*8 + i*4;
            float4 v = make_float4(tb[m0], tb[m0+1], tb[m0+2], tb[m0+3]);
            *(float4*)&outs[k][(size_t)(cB + ln) * W2 + rowBase + m0] = v;  // transposed store
        }
    }
}

// ---------------- pass 2: horizontal conv (WMMA) + fused eigen/vesselness ----------------
__global__ __launch_bounds__(256)
void k_hess(const float* __restrict__ pdd, const float* __restrict__ pds,
            const float* __restrict__ pss, const float* __restrict__ kbuf,
            int r, int mode, float* __restrict__ sqp, float* __restrict__ rbp,
            float* __restrict__ fm, unsigned* __restrict__ scal) {
    __shared__ float sIn[16 * SSTRIDE];
    __shared__ float skp[3 * KP];
    const int R = 2*r, SW = 128 + 4*r, chunks = r + 4;
    const int tid = threadIdx.x;
    const int wv = tid >> 5, lane = tid & 31, half = lane >> 4, ln = lane & 15;
    const int rowBase  = blockIdx.y * 16;
    const int colBase0 = blockIdx.x * 128;

    float gs_inv = 0.f;
    if (mode == 1) {
        float ms = __uint_as_float(scal[0]);
        float gs = (ms == 0.f) ? 2.f : 0.5f * ms;   // gamma_sq
        gs_inv = 1.f / gs;
    }

    for (int i = tid; i < 3*KP; i += 256) skp[i] = kbuf[i];

    const float* planes[3] = {pdd, pds, pss};
    const int ksel[3] = {2, 1, 0};   // hrr: ss, hrc: ds, hcc: dd
    v8f acc[3];
    #pragma unroll
    for (int p = 0; p < 3; ++p) {
        __syncthreads();
        for (int rr = 0; rr < 16; ++rr) {
            const float* src = planes[p] + (size_t)(rowBase + rr) * W2;
            for (int c = tid; c < SW; c += 256)
                sIn[rr*SSTRIDE + c] = src[reflect_idx(colBase0 - R + c, W2)];
        }
        __syncthreads();
        v8f a8 = {};
        const float* arow = &sIn[ln * SSTRIDE + wv*16 + 2*half];
        const float* kp   = &skp[ksel[p]*KP + 16 + 2*half - ln];
        for (int kc = 0; kc < chunks; ++kc) {
            v2f a; a.x = arow[kc*4]; a.y = arow[kc*4 + 1];
            v2f b; b.x = kp[kc*4];   b.y = kp[kc*4 + 1];
            a8 = __builtin_amdgcn_wmma_f32_16x16x4_f32(false, a, false, b, (short)0, a8, false, false);
        }
        acc[p] = a8;
    }

    const int colg = colBase0 + wv*16 + ln;
    float localmax = 0.f;
    #pragma unroll
    for (int vg = 0; vg < 8; ++vg) {
        float hrr = acc[0][vg], hrc = acc[1][vg], hcc = acc[2][vg];
        float ht = 0.5f * (hrr + hcc);
        float df = hrr - hcc;
        float disc = sqrtf(0.25f*df*df + hrc*hrc);
        float e1 = ht + disc, e2 = ht - disc;
        bool sw = fabsf(e1) > fabsf(e2);
        float lam1 = sw ? e2 : e1;     // smaller |.|
        float lam2 = sw ? e1 : e2;     // larger |.|
        float ssq = lam1*lam1 + lam2*lam2;
        float l2c = fmaxf(lam2, 1e-10f);
        float rb  = (lam1*lam1) / (l2c*l2c);
        size_t idx = (size_t)(rowBase + vg + 8*half) * W2 + colg;
        if (mode == 0) {
            sqp[idx] = ssq; rbp[idx] = rb;
            localmax = fmaxf(localmax, ssq);
        } else {
            float v = expf(-rb * (1.f/0.045f)) * (1.f - expf(-ssq * gs_inv));
            fm[idx] = fmaxf(fm[idx], v);
        }
    }
    if (mode == 0) atomicMax(&scal[0], __float_as_uint(localmax));
}

// ---------------- sigma0: vals from stored S^2, R_B^2 once gamma known ----------------
__global__ void k_vals0(float* __restrict__ fmsq, const float* __restrict__ rbp,
                        const unsigned* __restrict__ scal) {
    float ms = __uint_as_float(scal[0]);
    float gs = (ms == 0.f) ? 2.f : 0.5f * ms;
    float gi = 1.f / gs;
    int i = blockIdx.x * blockDim.x + threadIdx.x;
    float ssq = fmsq[i];
    float rb  = rbp[i];
    fmsq[i] = expf(-rb * (1.f/0.045f)) * (1.f - expf(-ssq * gi));
}

__global__ void k_init(unsigned* scal) {
    scal[0] = 0u;            // max S^2 (sigma0)
    scal[1] = 0x7f800000u;   // +inf  (min of fm)
    scal[2] = 0u;            // 0     (max of fm)
}

__global__ __launch_bounds__(256)
void k_minmax(const float* __restrict__ fm, unsigned* __restrict__ scal) {
    __shared__ float smn[256], smx[256];
    const int tid = threadIdx.x;
    float mn = 1e30f, mx = 0.f;
    for (int i = blockIdx.x*256 + tid; i < NPIX; i += gridDim.x*256) {
        float v = fm[i];
        mn = fminf(mn, v); mx = fmaxf(mx, v);
    }
    smn[tid] = mn; smx[tid] = mx;
    __syncthreads();
    for (int s = 128; s > 0; s >>= 1) {
        if (tid < s) {
            smn[tid] = fminf(smn[tid], smn[tid+s]);
            smx[tid] = fmaxf(smx[tid], smx[tid+s]);
        }
        __syncthreads();
    }
    if (tid == 0) {
        atomicMin(&scal[1], __float_as_uint(smn[0]));   // non-negative: uint order == float order
        atomicMax(&scal[2], __float_as_uint(smx[0]));
    }
}

__global__ void k_norm(float* __restrict__ fm, const unsigned* __restrict__ scal) {
    float mn = __uint_as_float(scal[1]);
    float mx = __uint_as_float(scal[2]);
    float d = mx - mn;
    float inv = (d > 0.f) ? 1.f/d : 0.f;
    int i = blockIdx.x * blockDim.x + threadIdx.x;
    fm[i] = (fm[i] - mn) * inv;
}

extern "C" void kernel_launch(void* const* d_in, const int* in_sizes, int n_in,
                              void* d_out, int out_size, void* d_ws, size_t ws_size,
                              hipStream_t stream) {
    const float* img = (const float*)d_in[0];
    float* out = (float*)d_out;
    float* ws  = (float*)d_ws;

    float* grayT = ws;
    float* pdd   = ws + (size_t)1*NPIX;
    float* pds   = ws + (size_t)2*NPIX;
    float* pss   = ws + (size_t)3*NPIX;
    float* rbp   = ws + (size_t)4*NPIX;
    float* kbuf  = ws + (size_t)5*NPIX;
    unsigned* scal = (unsigned*)(ws + (size_t)5*NPIX + 4096);

    k_init<<<1, 1, 0, stream>>>(scal);
    k_gray_t<<<dim3(64, 64), dim3(32, 8), 0, stream>>>(img, grayT);

    const int sigmas[5] = {10, 12, 14, 16, 18};
    for (int s = 0; s < 5; ++s) {
        const int sigma = sigmas[s];
        const int r = 4 * sigma;                 // int(4*sigma + 0.5), sigma integer
        k_make_kernels<<<1, 512, 0, stream>>>(kbuf, sigma, r);
        k_conv_v<<<dim3(16, 128), 256, 0, stream>>>(grayT, kbuf, pdd, pds, pss, r);
        const int mode = (s == 0) ? 0 : 1;
        // mode0: S^2 -> d_out, R_B^2 -> rbp, atomic max S^2; mode1: fm(d_out) = max(fm, vals)
        k_hess<<<dim3(16, 128), 256, 0, stream>>>(pdd, pds, pss, kbuf, r, mode,
                                                  out, rbp, out, scal);
        if (s == 0)
            k_vals0<<<NPIX/256, 256, 0, stream>>>(out, rbp, scal);
    }
    k_minmax<<<1024, 256, 0, stream>>>(out, scal);
    k_norm<<<NPIX/256, 256, 0, stream>>>(out, scal);
}